// ResidueEmbedding_23046794510930
// MI455X (gfx1250) — compile-verified
//
#include <hip/hip_runtime.h>
#include <hip/hip_bf16.h>

// ---------------------------------------------------------------------------
// ResidueEmbedding for MI455X (gfx1250, wave32, WMMA).
//   Layer 1: rows binned by amino type; each block owns <=32 same-type rows
//            and caches the 75-row coord weight slice + 39 dih weights in
//            VGPRs (cuts L2 weight traffic ~32x, from 1.26 GB to ~59 MB).
//   Layers 2-4: fused, V_WMMA_F32_16X16X4_F32, 32 rows per wave so each
//            B fragment feeds two A tiles (16 WMMA per 18 VMEM).
// ---------------------------------------------------------------------------

typedef float v2f __attribute__((ext_vector_type(2)));
typedef float v8f __attribute__((ext_vector_type(8)));

#define NROWS   16384   // B*L
#define NATOM   25
#define TTYPES  20
#define NCHAIN  10
#define DF      128
#define H1W     256     // 2*DF
#define DIHW    39
#define COORDW  75      // NATOM*3
#define W1_COORD_OFF 128
#define W1_DIH_OFF   1628
#define W1_CHAIN_OFF 1667
#define MAXB    NROWS   // per-type bucket stride (worst case: all rows one type)
#define ROWS_PER_BLK 32

// ---------------------------------------------------------------------------
// Kernel A: gather tables  Taa[t] = aa_emb[t] @ W1[0:128]
//                          Tch[c] = chain_emb[c] @ W1[1667:1795]
// ---------------------------------------------------------------------------
__global__ void __launch_bounds__(256)
tables_kernel(const float* __restrict__ aa_emb, const float* __restrict__ chain_emb,
              const float* __restrict__ W1, float* __restrict__ Taa,
              float* __restrict__ Tch)
{
    const int col = threadIdx.x;
    const int b   = blockIdx.x;           // 0..29
    if (b < TTYPES) {
        float acc = 0.f;
        for (int k = 0; k < DF; ++k)
            acc = fmaf(aa_emb[b * DF + k], W1[k * H1W + col], acc);
        Taa[b * H1W + col] = acc;
    } else {
        const int c = b - TTYPES;
        float acc = 0.f;
        if (c != 0) {
            for (int k = 0; k < DF; ++k)
                acc = fmaf(chain_emb[c * DF + k], W1[(W1_CHAIN_OFF + k) * H1W + col], acc);
        }
        Tch[c * H1W + col] = acc;
    }
}

// ---------------------------------------------------------------------------
// Binning: cursors[t] starts at t*MAXB; scatter row ids into padded buckets.
// Output is order-independent (each row computes into its own h1 slot).
// ---------------------------------------------------------------------------
__global__ void __launch_bounds__(32)
init_cursors_kernel(int* __restrict__ cursors)
{
    const int t = threadIdx.x;
    if (t < TTYPES) cursors[t] = t * MAXB;
}

__global__ void __launch_bounds__(256)
scatter_kernel(const int* __restrict__ seq, int* __restrict__ cursors,
               int* __restrict__ bucket, int n)
{
    const int i = blockIdx.x * 256 + threadIdx.x;
    if (i < n) {
        const int t   = seq[i];
        const int pos = atomicAdd(&cursors[t], 1);
        bucket[pos] = i;
    }
}

// ---------------------------------------------------------------------------
// Kernel B: layer 1, binned. block(y=t, x=j) handles rows
//   bucket[t*MAXB + j*32 ... +nr).  Thread = output column; the 75 coord
//   weights (type t) + 39 dih weights live in VGPRs, reused across 32 rows.
//   Activations staged in LDS, read as wave-uniform broadcasts.
// ---------------------------------------------------------------------------
__global__ void __launch_bounds__(256)
layer1_binned_kernel(const int* __restrict__ bucket, const int* __restrict__ cursors,
                     const float* __restrict__ xyz, const float* __restrict__ dihedrals,
                     const int* __restrict__ chain_idx, const float* __restrict__ orient,
                     const float* __restrict__ W1, const float* __restrict__ b1,
                     const float* __restrict__ Taa, const float* __restrict__ Tch,
                     float* __restrict__ h1)
{
    const int t     = blockIdx.y;                  // amino type 0..19
    const int start = t * MAXB + blockIdx.x * ROWS_PER_BLK;
    const int end   = cursors[t];                  // t*MAXB + count[t]
    if (start >= end) return;
    const int nr = min(ROWS_PER_BLK, end - start);

    __shared__ float s_x[ROWS_PER_BLK][COORDW + 1];
    __shared__ float s_dih[ROWS_PER_BLK][DIHW + 1];
    __shared__ int   s_row[ROWS_PER_BLK];

    const int tid = threadIdx.x;
    if (tid < nr) s_row[tid] = bucket[start + tid];
    __syncthreads();

    // local-frame coords: out[a][i] = sum_j O[j][i]*(x[a][j]-x_CA[j])
    for (int idx = tid; idx < nr * NATOM; idx += 256) {
        const int r = idx / NATOM, a = idx % NATOM;
        const int row = s_row[r];
        const float* X = xyz    + (size_t)row * COORDW;
        const float* R = orient + (size_t)row * 9;
        const float rx = X[a * 3 + 0] - X[3];
        const float ry = X[a * 3 + 1] - X[4];
        const float rz = X[a * 3 + 2] - X[5];
        s_x[r][a * 3 + 0] = R[0] * rx + R[3] * ry + R[6] * rz;
        s_x[r][a * 3 + 1] = R[1] * rx + R[4] * ry + R[7] * rz;
        s_x[r][a * 3 + 2] = R[2] * rx + R[5] * ry + R[8] * rz;
    }
    // angular encoding: per dihedral d: [x, sin(f_i x)*6, cos(f_i x)*6]
    for (int idx = tid; idx < nr * DIHW; idx += 256) {
        const int r = idx / DIHW, k = idx % DIHW;
        const int d = k / 13, s = k % 13;
        const float freqs[6] = {1.f, 2.f, 3.f, 1.f, 0.5f, 1.f / 3.f};
        const float x = dihedrals[(size_t)s_row[r] * 3 + d];
        float v;
        if (s == 0)      v = x;
        else if (s < 7)  v = sinf(freqs[s - 1] * x);
        else             v = cosf(freqs[s - 7] * x);
        s_dih[r][k] = v;
    }
    __syncthreads();

    const int col = tid;                           // 256 output columns
    float wc[COORDW];
    {
        const float* Wc = W1 + (size_t)(W1_COORD_OFF + t * COORDW) * H1W + col;
        #pragma unroll
        for (int k = 0; k < COORDW; ++k) wc[k] = Wc[(size_t)k * H1W];
    }
    float wd[DIHW];
    {
        const float* Wd = W1 + (size_t)W1_DIH_OFF * H1W + col;
        #pragma unroll
        for (int k = 0; k < DIHW; ++k) wd[k] = Wd[(size_t)k * H1W];
    }
    const float base = b1[col] + Taa[t * H1W + col];

    for (int r = 0; r < nr; ++r) {
        const int row = s_row[r];
        float acc = base + Tch[chain_idx[row] * H1W + col];
        #pragma unroll
        for (int k = 0; k < COORDW; ++k) acc = fmaf(s_x[r][k], wc[k], acc);
        #pragma unroll
        for (int k = 0; k < DIHW; ++k)   acc = fmaf(s_dih[r][k], wd[k], acc);
        h1[(size_t)row * H1W + col] = acc > 0.f ? acc : 0.f;
    }
}

// ---------------------------------------------------------------------------
// Kernel C: fused layers 2-4 with V_WMMA_F32_16X16X4_F32.
//   Block = 64 threads = 2 waves; each wave owns 32 rows (two 16-row A
//   tiles) so every B fragment feeds two WMMAs.  One per-wave 32x132 LDS
//   slab is reused for h2 then h3 (in-wave DS ordering makes this safe).
// ---------------------------------------------------------------------------
#define S3 132   // padded LDS row stride (floats); 132 % 64 = 4 spreads banks

__device__ __forceinline__ void zero8(v8f& v) {
    #pragma unroll
    for (int i = 0; i < 8; ++i) v[i] = 0.f;
}

__global__ void __launch_bounds__(64)
mlp_tail_kernel(const float* __restrict__ h1,
                const float* __restrict__ W2, const float* __restrict__ b2,
                const float* __restrict__ W3, const float* __restrict__ b3,
                const float* __restrict__ W4, const float* __restrict__ b4,
                float* __restrict__ out)
{
    __shared__ float lds[2 * 32 * S3];             // per-wave 32 x S3 slab

    const int tid    = threadIdx.x;
    const int w      = tid >> 5;
    const int lane   = tid & 31;
    const int row_lo = lane & 15;                  // M within a 16-row tile
    const int koff   = (lane < 16) ? 0 : 2;        // A/B fragment K split
    const int hi8    = (lane < 16) ? 0 : 8;        // C fragment row split
    const int m0     = blockIdx.x * 64 + w * 32;   // first of this wave's 32 rows

    float* my = lds + w * (32 * S3);

    v8f acc[2][8];                                 // [row tile][col tile]

    // ---------------- layer 2: (32x256) @ (256x128), relu ----------------
    #pragma unroll
    for (int i = 0; i < 2; ++i)
        #pragma unroll
        for (int n = 0; n < 8; ++n) zero8(acc[i][n]);

    const float* ap0 = h1 + (size_t)(m0 + row_lo) * H1W + koff;
    const float* ap1 = h1 + (size_t)(m0 + 16 + row_lo) * H1W + koff;
    for (int k0 = 0; k0 < H1W; k0 += 4) {
        const v2f a0 = *(const v2f*)(ap0 + k0);    // A: VGPR0=K{0,2}, VGPR1=K{1,3}
        const v2f a1 = *(const v2f*)(ap1 + k0);
        #pragma unroll
        for (int n = 0; n < 8; ++n) {
            const int nn = n * 16 + row_lo;
            v2f bf;
            bf[0] = W2[(size_t)(k0 + koff)     * DF + nn];
            bf[1] = W2[(size_t)(k0 + koff + 1) * DF + nn];
            acc[0][n] = __builtin_amdgcn_wmma_f32_16x16x4_f32(
                            false, a0, false, bf, (short)0, acc[0][n], false, false);
            acc[1][n] = __builtin_amdgcn_wmma_f32_16x16x4_f32(
                            false, a1, false, bf, (short)0, acc[1][n], false, false);
        }
    }
    #pragma unroll
    for (int i = 0; i < 2; ++i)
        #pragma unroll
        for (int n = 0; n < 8; ++n) {
            const int nn = n * 16 + row_lo;
            const float bb = b2[nn];
            #pragma unroll
            for (int r = 0; r < 8; ++r) {
                float v = acc[i][n][r] + bb;
                my[(i * 16 + r + hi8) * S3 + nn] = v > 0.f ? v : 0.f;
            }
        }

    // ---------------- layer 3: (32x128) @ (128x128), relu ----------------
    #pragma unroll
    for (int i = 0; i < 2; ++i)
        #pragma unroll
        for (int n = 0; n < 8; ++n) zero8(acc[i][n]);

    for (int k0 = 0; k0 < DF; k0 += 4) {
        const v2f a0 = *(const v2f*)(my + row_lo * S3 + k0 + koff);
        const v2f a1 = *(const v2f*)(my + (16 + row_lo) * S3 + k0 + koff);
        #pragma unroll
        for (int n = 0; n < 8; ++n) {
            const int nn = n * 16 + row_lo;
            v2f bf;
            bf[0] = W3[(size_t)(k0 + koff)     * DF + nn];
            bf[1] = W3[(size_t)(k0 + koff + 1) * DF + nn];
            acc[0][n] = __builtin_amdgcn_wmma_f32_16x16x4_f32(
                            false, a0, false, bf, (short)0, acc[0][n], false, false);
            acc[1][n] = __builtin_amdgcn_wmma_f32_16x16x4_f32(
                            false, a1, false, bf, (short)0, acc[1][n], false, false);
        }
    }
    // overwrite the slab with h3 (all h2 reads are program-order earlier)
    #pragma unroll
    for (int i = 0; i < 2; ++i)
        #pragma unroll
        for (int n = 0; n < 8; ++n) {
            const int nn = n * 16 + row_lo;
            const float bb = b3[nn];
            #pragma unroll
            for (int r = 0; r < 8; ++r) {
                float v = acc[i][n][r] + bb;
                my[(i * 16 + r + hi8) * S3 + nn] = v > 0.f ? v : 0.f;
            }
        }

    // ---------------- layer 4: (32x128) @ (128x128), no relu --------------
    #pragma unroll
    for (int i = 0; i < 2; ++i)
        #pragma unroll
        for (int n = 0; n < 8; ++n) zero8(acc[i][n]);

    for (int k0 = 0; k0 < DF; k0 += 4) {
        const v2f a0 = *(const v2f*)(my + row_lo * S3 + k0 + koff);
        const v2f a1 = *(const v2f*)(my + (16 + row_lo) * S3 + k0 + koff);
        #pragma unroll
        for (int n = 0; n < 8; ++n) {
            const int nn = n * 16 + row_lo;
            v2f bf;
            bf[0] = W4[(size_t)(k0 + koff)     * DF + nn];
            bf[1] = W4[(size_t)(k0 + koff + 1) * DF + nn];
            acc[0][n] = __builtin_amdgcn_wmma_f32_16x16x4_f32(
                            false, a0, false, bf, (short)0, acc[0][n], false, false);
            acc[1][n] = __builtin_amdgcn_wmma_f32_16x16x4_f32(
                            false, a1, false, bf, (short)0, acc[1][n], false, false);
        }
    }
    #pragma unroll
    for (int i = 0; i < 2; ++i)
        #pragma unroll
        for (int n = 0; n < 8; ++n) {
            const int nn = n * 16 + row_lo;
            const float bb = b4[nn];
            #pragma unroll
            for (int r = 0; r < 8; ++r)
                out[(size_t)(m0 + i * 16 + r + hi8) * DF + nn] = acc[i][n][r] + bb;
        }
}

// ---------------------------------------------------------------------------
extern "C" void kernel_launch(void* const* d_in, const int* in_sizes, int n_in,
                              void* d_out, int out_size, void* d_ws, size_t ws_size,
                              hipStream_t stream)
{
    (void)n_in; (void)out_size; (void)ws_size;
    const int*   seq       = (const int*)  d_in[0];
    const float* xyz       = (const float*)d_in[1];
    const float* dihedrals = (const float*)d_in[2];
    const int*   chain_idx = (const int*)  d_in[3];
    const float* orient    = (const float*)d_in[4];
    /* d_in[5] atom_mask: unused by the reference */
    const float* aa_emb    = (const float*)d_in[6];
    const float* chain_emb = (const float*)d_in[7];
    const float* W1 = (const float*)d_in[8];
    const float* b1 = (const float*)d_in[9];
    const float* W2 = (const float*)d_in[10];
    const float* b2 = (const float*)d_in[11];
    const float* W3 = (const float*)d_in[12];
    const float* b3 = (const float*)d_in[13];
    const float* W4 = (const float*)d_in[14];
    const float* b4 = (const float*)d_in[15];

    const int N = in_sizes[0];                 // B*L = 16384 rows

    // ws layout (fp32/int32):
    //   Taa[20*256] | Tch[10*256] | h1[N*256] | cursors[32] | bucket[20*MAXB]
    float* Taa = (float*)d_ws;
    float* Tch = Taa + TTYPES * H1W;
    float* h1  = Tch + NCHAIN * H1W;
    int*   cursors = (int*)(h1 + (size_t)N * H1W);
    int*   bucket  = cursors + 32;

    tables_kernel<<<TTYPES + NCHAIN, 256, 0, stream>>>(aa_emb, chain_emb, W1, Taa, Tch);
    init_cursors_kernel<<<1, 32, 0, stream>>>(cursors);
    scatter_kernel<<<(N + 255) / 256, 256, 0, stream>>>(seq, cursors, bucket, N);

    dim3 g1((N + ROWS_PER_BLK - 1) / ROWS_PER_BLK, TTYPES);
    layer1_binned_kernel<<<g1, 256, 0, stream>>>(bucket, cursors, xyz, dihedrals,
                                                 chain_idx, orient, W1, b1,
                                                 Taa, Tch, h1);

    mlp_tail_kernel<<<N / 64, 64, 0, stream>>>(h1, W2, b2, W3, b3, W4, b4,
                                               (float*)d_out);
}